// KVAttnDecoderRNN_79517024518151
// MI455X (gfx1250) — compile-verified
//
#include <hip/hip_runtime.h>
#include <hip/hip_bf16.h>
#include <math.h>

// Sizes (compile-time constants from the reference)
#define VOCAB 32000
#define H     512
#define B     10
#define KB    431
#define KB_PAD 1523
#define KB_TOT (KB_PAD + KB)   // 1954

typedef __attribute__((ext_vector_type(2))) float v2f;
typedef __attribute__((ext_vector_type(8))) float v8f;
typedef __attribute__((address_space(3))) float lds_f;

__device__ __forceinline__ float sigmoidf_(float x) { return 1.0f / (1.0f + __expf(-x)); }

// ---------------------------------------------------------------------------
// Kernel 1: embedding lookup + single GRU step -> h1 (written to hidden slot)
// ---------------------------------------------------------------------------
__global__ void k_gru(const int* __restrict__ input_seq,
                      const float* __restrict__ last_hidden,   // (1,B,H)
                      const float* __restrict__ emb,           // (VOCAB,H)
                      const float* __restrict__ w_ih,          // (3H,H)
                      const float* __restrict__ w_hh,          // (3H,H)
                      const float* __restrict__ b_ih,
                      const float* __restrict__ b_hh,
                      float* __restrict__ h1_out)              // (B,H)
{
    int t = blockIdx.x * blockDim.x + threadIdx.x;   // 0..5119
    int b = t >> 9;
    int j = t & (H - 1);

    const float* x  = emb + (size_t)input_seq[b] * H;
    const float* h0 = last_hidden + b * H;

    float gir = b_ih[j], giz = b_ih[H + j], gin = b_ih[2 * H + j];
    float ghr = b_hh[j], ghz = b_hh[H + j], ghn = b_hh[2 * H + j];

    const float* wri = w_ih + (size_t)j * H;
    const float* wzi = w_ih + (size_t)(H + j) * H;
    const float* wni = w_ih + (size_t)(2 * H + j) * H;
    const float* wrh = w_hh + (size_t)j * H;
    const float* wzh = w_hh + (size_t)(H + j) * H;
    const float* wnh = w_hh + (size_t)(2 * H + j) * H;

    for (int h = 0; h < H; ++h) {
        float xv = x[h], hv = h0[h];
        gir = fmaf(xv, wri[h], gir);
        giz = fmaf(xv, wzi[h], giz);
        gin = fmaf(xv, wni[h], gin);
        ghr = fmaf(hv, wrh[h], ghr);
        ghz = fmaf(hv, wzh[h], ghz);
        ghn = fmaf(hv, wnh[h], ghn);
    }
    float r = sigmoidf_(gir + ghr);
    float z = sigmoidf_(giz + ghz);
    float n = tanhf(gin + r * ghn);
    h1_out[t] = (1.0f - z) * n + z * h0[j];
}

// ---------------------------------------------------------------------------
// Kernel 2: energies[b,l] = <h1[b,:], enc[l,b,:]>   (one wave per output)
// ---------------------------------------------------------------------------
__global__ void k_energies(const float* __restrict__ h1,    // (B,H)
                           const float* __restrict__ enc,   // (H,B,H)
                           float* __restrict__ energies)    // (B,L)
{
    int gid  = blockIdx.x * blockDim.x + threadIdx.x;
    int wid  = gid >> 5;                 // 0..5119
    int lane = threadIdx.x & 31;
    int b = wid >> 9;
    int l = wid & (H - 1);

    const float* hv = h1 + b * H;
    const float* ev = enc + ((size_t)l * B + b) * H;
    float s = 0.0f;
    for (int h = lane; h < H; h += 32) s = fmaf(hv[h], ev[h], s);
    #pragma unroll
    for (int off = 16; off > 0; off >>= 1) s += __shfl_xor(s, off, 32);
    if (lane == 0) energies[wid] = s;
}

// ---------------------------------------------------------------------------
// Kernel 3: softmax over the BATCH axis (axis=0) per column l -> attn (B,L)
// ---------------------------------------------------------------------------
__global__ void k_softmax(const float* __restrict__ energies,  // (B,L)
                          float* __restrict__ attn)            // (B,1,L)
{
    int l = blockIdx.x * blockDim.x + threadIdx.x;
    if (l >= H) return;
    float m = -INFINITY;
    #pragma unroll
    for (int b = 0; b < B; ++b) m = fmaxf(m, energies[b * H + l]);
    float ex[B];
    float sum = 0.0f;
    #pragma unroll
    for (int b = 0; b < B; ++b) { ex[b] = __expf(energies[b * H + l] - m); sum += ex[b]; }
    float inv = 1.0f / sum;
    #pragma unroll
    for (int b = 0; b < B; ++b) attn[b * H + l] = ex[b] * inv;
}

// ---------------------------------------------------------------------------
// Kernel 4: context[b,h] = sum_l attn[b,l] * enc[l,b,h]
// ---------------------------------------------------------------------------
__global__ void k_context(const float* __restrict__ attn,   // (B,L)
                          const float* __restrict__ enc,    // (L,B,H)
                          float* __restrict__ ctx)          // (B,H)
{
    int t = blockIdx.x * blockDim.x + threadIdx.x;  // 0..5119
    int b = t >> 9;
    int h = t & (H - 1);
    const float* ab = attn + b * H;
    float s = 0.0f;
    for (int l = 0; l < H; ++l)
        s = fmaf(ab[l], enc[((size_t)l * B + b) * H + h], s);
    ctx[t] = s;
}

// ---------------------------------------------------------------------------
// Kernel 5: concat_pad[m][k] = tanh([h1|ctx] @ w_concat.T + b_concat),
//           zero-padded to 16 rows (WMMA A-matrix staging, EXEC stays full)
// ---------------------------------------------------------------------------
__global__ void k_concat(const float* __restrict__ h1,
                         const float* __restrict__ ctx,
                         const float* __restrict__ w_concat,  // (H, 2H)
                         const float* __restrict__ b_concat,
                         float* __restrict__ cpad)            // (16, H)
{
    int t = blockIdx.x * blockDim.x + threadIdx.x;  // 0..8191
    int b = t >> 9;
    int j = t & (H - 1);
    if (b >= B) { cpad[t] = 0.0f; return; }
    float s = b_concat[j];
    const float* w  = w_concat + (size_t)j * (2 * H);
    const float* hb = h1 + b * H;
    const float* cb = ctx + b * H;
    for (int h = 0; h < H; ++h) {
        s = fmaf(hb[h], w[h], s);
        s = fmaf(cb[h], w[H + h], s);
    }
    cpad[t] = tanhf(s);
}

// ---------------------------------------------------------------------------
// Kernel 6: output(10 x 32000) = cpad(16 x 512) @ w_out^T + b_out
// via V_WMMA_F32_16X16X4_F32.
//  - A tile (16x512 f32 = 32 KB) is async-copied into LDS once per block
//    (global_load_async_to_lds_b128, ASYNCcnt), then served by the DS pipe
//    so the VMEM pipe is dedicated to streaming w_out (65.5 MB).
//  - One wave per 16-column tile of N; 128 WMMA per wave; C seeded with bias.
// ---------------------------------------------------------------------------
__global__ void k_outgemm(const float* __restrict__ cpad,   // (16, H)
                          const float* __restrict__ w_out,  // (VOCAB, H)
                          const float* __restrict__ b_out,  // (VOCAB)
                          float* __restrict__ out)          // (B, VOCAB)
{
    __shared__ float lds_a[16 * H];               // 32 KB A tile

    int tid  = threadIdx.x;                       // 0..127
    int wave = tid >> 5;                          // 0..3
    int lane = tid & 31;

    // --- Cooperative async copy: cpad -> LDS, 16 B per lane per step -------
    {
        lds_f* lbase = (lds_f*)lds_a;
        #pragma unroll
        for (int it = 0; it < 16; ++it) {
            int elt = it * 128 + tid;             // b128 chunk index, 0..2047
            lds_f* ldst       = lbase + elt * 4;  // 32-bit LDS byte offset in VGPR
            const float* gsrc = cpad + (size_t)elt * 4;
            asm volatile("global_load_async_to_lds_b128 %0, %1, off"
                         :: "v"(ldst), "v"(gsrc) : "memory");
        }
        asm volatile("s_wait_asynccnt 0x0" ::: "memory");
    }
    __syncthreads();

    int tile = blockIdx.x * 4 + wave;             // 0..1999
    int n0   = tile * 16;
    int m16  = lane & 15;
    int hi   = lane >> 4;
    int n    = n0 + m16;

    float bias = b_out[n];
    v8f c = {bias, bias, bias, bias, bias, bias, bias, bias};

    const float* arow = &lds_a[m16 * H];          // LDS-backed A row
    const float* brow = w_out + (size_t)n * H;

    for (int k0 = 0; k0 < H; k0 += 4) {
        int kp = k0 + 2 * hi;
        __builtin_prefetch(brow + kp + 64, 0, 1); // global_prefetch_b8 on w_out stream
        v2f a  = *(const v2f*)(arow + kp);        // ds_load_b64
        v2f bb = *(const v2f*)(brow + kp);        // global_load_b64
        // D = A(16x4) * B(4x16) + C
        c = __builtin_amdgcn_wmma_f32_16x16x4_f32(false, a, false, bb,
                                                  (short)0, c, false, false);
    }

    #pragma unroll
    for (int r = 0; r < 8; ++r) {
        int m = r + 8 * hi;
        if (m < B) out[(size_t)m * VOCAB + n] = c[r];
    }
}

// ---------------------------------------------------------------------------
// Kernel 7: kb_attn (B, 512, 1954): zeros prefix + reshaped 3-way gather sum
// ---------------------------------------------------------------------------
__global__ void k_kb(const int* __restrict__ kb_inputs,     // (B, KB, 3)
                     const float* __restrict__ emb_kb,      // (VOCAB, H)
                     float* __restrict__ out_kb)            // (B, 512, 1954)
{
    int t = blockIdx.x * blockDim.x + threadIdx.x;
    if (t >= B * H * KB_TOT) return;
    int c  = t % KB_TOT;
    int bp = t / KB_TOT;
    int p  = bp & (H - 1);
    int b  = bp >> 9;

    float v = 0.0f;
    if (c >= KB_PAD) {
        int i  = p * KB + (c - KB_PAD);
        int kk = i >> 9;
        int h  = i & (H - 1);
        const int* kin = kb_inputs + ((size_t)b * KB + kk) * 3;
        v = emb_kb[(size_t)kin[0] * H + h]
          + emb_kb[(size_t)kin[1] * H + h]
          + emb_kb[(size_t)kin[2] * H + h];
    }
    out_kb[t] = v;
}

// ---------------------------------------------------------------------------
extern "C" void kernel_launch(void* const* d_in, const int* in_sizes, int n_in,
                              void* d_out, int out_size, void* d_ws, size_t ws_size,
                              hipStream_t stream) {
    const int*   input_seq   = (const int*)  d_in[0];
    const int*   kb_inputs   = (const int*)  d_in[1];
    /* d_in[2] = last_context : unused by the reference */
    const float* last_hidden = (const float*)d_in[3];
    const float* enc         = (const float*)d_in[4];
    const float* emb         = (const float*)d_in[5];
    const float* emb_kb      = (const float*)d_in[6];
    const float* w_ih        = (const float*)d_in[7];
    const float* w_hh        = (const float*)d_in[8];
    const float* b_ih        = (const float*)d_in[9];
    const float* b_hh        = (const float*)d_in[10];
    const float* w_concat    = (const float*)d_in[11];
    const float* b_concat    = (const float*)d_in[12];
    const float* w_out       = (const float*)d_in[13];
    const float* b_out       = (const float*)d_in[14];

    // Output tuple layout (flat, return order):
    float* out        = (float*)d_out;
    float* out_output = out;                        // (B, VOCAB)  320000
    float* out_ctx    = out + B * VOCAB;            // (B, H)        5120
    float* out_hid    = out_ctx + B * H;            // (1, B, H)     5120
    float* out_attn   = out_hid + B * H;            // (B, 1, H)     5120
    float* out_kb     = out_attn + B * H;           // (B, H, 1954)

    float* ws       = (float*)d_ws;
    float* energies = ws;                 // 5120 floats
    float* cpad     = ws + B * H;         // 16*512 floats, 16B-aligned

    k_gru     <<<(B * H) / 256, 256, 0, stream>>>(input_seq, last_hidden, emb,
                                                  w_ih, w_hh, b_ih, b_hh, out_hid);
    k_energies<<<(B * H * 32) / 256, 256, 0, stream>>>(out_hid, enc, energies);
    k_softmax <<<2, 256, 0, stream>>>(energies, out_attn);
    k_context <<<(B * H) / 256, 256, 0, stream>>>(out_attn, enc, out_ctx);
    k_concat  <<<(16 * H) / 256, 256, 0, stream>>>(out_hid, out_ctx,
                                                   w_concat, b_concat, cpad);
    k_outgemm <<<(VOCAB / 16) / 4, 128, 0, stream>>>(cpad, w_out, b_out, out_output);
    k_kb      <<<(B * H * KB_TOT + 255) / 256, 256, 0, stream>>>(kb_inputs, emb_kb, out_kb);
}